// TorchWrapperWithMetrics_89850715833247
// MI455X (gfx1250) — compile-verified
//
#include <hip/hip_runtime.h>
#include <hip/hip_bf16.h>
#include <math.h>

typedef float v2f __attribute__((ext_vector_type(2)));
typedef float v8f __attribute__((ext_vector_type(8)));

#define KNN          20
#define DFEAT        64
#define BQ           2048
#define NPTS         100000
#define HDIM         256
#define QTILE        32
#define NCHUNK       16
#define NSLICES      32
#define SLICE_GROUPS 4
#define WPB          8                          // waves per block (256 threads)
#define TOTAL_CHUNKS (NPTS / NCHUNK)            // 6250 (exact)
#define CPS ((TOTAL_CHUNKS + NSLICES - 1) / NSLICES)  // 196 chunks per slice
#define DT_STRIDE    20                         // dist-tile row stride (80B: 16B-aligned rows)

// ---------------------------------------------------------------------------
// Kernel 1: per-point squared norms of `data`
// ---------------------------------------------------------------------------
__global__ void dnorm_kernel(const float* __restrict__ data, float* __restrict__ dnorm) {
    int i = blockIdx.x * blockDim.x + threadIdx.x;
    if (i >= NPTS) return;
    const float* r = data + (size_t)i * DFEAT;
    float s = 0.f;
    #pragma unroll
    for (int c = 0; c < DFEAT; ++c) { float v = r[c]; s += v * v; }
    dnorm[i] = s;
}

// ---------------------------------------------------------------------------
// Kernel 2: fused WMMA distance GEMM + streaming per-slice top-20
// grid = 64 qblocks * 4 slice groups; block = 256 (8 waves)
// wave (qblock, slice) scans chunks [slice*CPS, min(6250,(slice+1)*CPS))
// A operands hoisted to registers; B chunk buffered in registers per chunk.
// ---------------------------------------------------------------------------
__global__ __launch_bounds__(256)
void knn_partial_kernel(const float* __restrict__ z, const float* __restrict__ data,
                        const float* __restrict__ dnorm,
                        float* __restrict__ pdist, int* __restrict__ pidx)
{
    __shared__ float xs[QTILE][66];      // 32 queries x 64 dims
    __shared__ float xn_s[QTILE];        // query squared norms
    __shared__ __align__(16) float dt[WPB][QTILE][DT_STRIDE]; // per-wave 32x16 dist tile

    const int tid    = threadIdx.x;
    const int wave   = tid >> 5;
    const int lane   = tid & 31;
    const int qblock = blockIdx.x / SLICE_GROUPS;
    const int sgroup = blockIdx.x % SLICE_GROUPS;
    const int slice  = sgroup * WPB + wave;

    // cooperative load of the 32x64 query tile from z (row stride 66)
    for (int i = tid; i < QTILE * DFEAT; i += 256) {
        int r = i >> 6, c = i & 63;
        xs[r][c] = z[(size_t)(qblock * QTILE + r) * (DFEAT + 2) + c];
    }
    __syncthreads();
    if (tid < QTILE) {
        float s = 0.f;
        #pragma unroll
        for (int c = 0; c < DFEAT; ++c) { float v = xs[tid][c]; s += v * v; }
        xn_s[tid] = s;
    }
    __syncthreads();

    const int n    = lane & 15;  // B column / C column (N)
    const int half = lane >> 4;  // K-half select for A/B, M-half select for C

    // hoist loop-invariant A operands and query norms into registers
    v2f a0r[16], a1r[16];
    #pragma unroll
    for (int kk = 0; kk < 16; ++kk) {
        const int kb = kk * 4 + half * 2;                  // lanes 0-15: K{0,1}; 16-31: K{2,3}
        a0r[kk].x = xs[n][kb];      a0r[kk].y = xs[n][kb + 1];
        a1r[kk].x = xs[16 + n][kb]; a1r[kk].y = xs[16 + n][kb + 1];
    }
    float xn0[8], xn1[8];
    #pragma unroll
    for (int v = 0; v < 8; ++v) { xn0[v] = xn_s[half * 8 + v]; xn1[v] = xn_s[16 + half * 8 + v]; }

    // register-resident unsorted top-20 with running max
    float kd[KNN]; int ki[KNN];
    #pragma unroll
    for (int i = 0; i < KNN; ++i) { kd[i] = 3.0e38f; ki[i] = 0; }
    float kmax = 3.0e38f; int kpos = 0;

    auto consider = [&](float d, int idx) {
        if (d < kmax) {
            #pragma unroll
            for (int i = 0; i < KNN; ++i) {                // replace current max (register-safe)
                bool rep = (i == kpos);
                kd[i] = rep ? d   : kd[i];
                ki[i] = rep ? idx : ki[i];
            }
            kmax = kd[0]; kpos = 0;
            #pragma unroll
            for (int i = 1; i < KNN; ++i) { if (kd[i] > kmax) { kmax = kd[i]; kpos = i; } }
        }
    };

    const int ch0 = slice * CPS;
    const int ch1 = min(TOTAL_CHUNKS, ch0 + CPS);

    for (int ch = ch0; ch < ch1; ++ch) {
        const int colbase = ch * NCHUNK;
        const int dpt     = colbase + n;                       // data point for this lane's B column
        const float* drow = data + (size_t)dpt * DFEAT;
        // unconditional speculative prefetch of next chunk (OOB prefetch is dropped by HW)
        __builtin_prefetch(drow + NCHUNK * DFEAT, 0, 1);

        // buffer the whole B chunk in registers first -> loads can be claused
        // and overlapped instead of load->wait->wmma serialization
        v2f br[16];
        #pragma unroll
        for (int kk = 0; kk < 16; ++kk) {
            const int kb = kk * 4 + half * 2;
            br[kk].x = drow[kb]; br[kk].y = drow[kb + 1];
        }

        v8f acc0 = {}; v8f acc1 = {};
        #pragma unroll
        for (int kk = 0; kk < 16; ++kk) {
            acc0 = __builtin_amdgcn_wmma_f32_16x16x4_f32(false, a0r[kk], false, br[kk],
                                                         (short)0, acc0, false, false);
            acc1 = __builtin_amdgcn_wmma_f32_16x16x4_f32(false, a1r[kk], false, br[kk],
                                                         (short)0, acc1, false, false);
        }

        // squared distances into per-wave LDS tile (transpose for row-wise scan)
        const float dn = dnorm[dpt];
        #pragma unroll
        for (int v = 0; v < 8; ++v) {
            const int r0 = half * 8 + v;                       // C layout: M = v + 8*half
            dt[wave][r0][n]      = xn0[v] + dn - 2.0f * acc0[v];
            dt[wave][16 + r0][n] = xn1[v] + dn - 2.0f * acc1[v];
        }
        asm volatile("s_wait_dscnt 0" ::: "memory");           // per-wave DS write->read ordering

        // each lane scans its own query row with 4x ds_load_b128;
        // quad-level min pre-check keeps the common path at 1 branch per 4 candidates
        const float4* rowv = (const float4*)(&dt[wave][lane][0]);
        #pragma unroll
        for (int j4 = 0; j4 < 4; ++j4) {
            float4 dv = rowv[j4];
            float m4 = fminf(fminf(dv.x, dv.y), fminf(dv.z, dv.w));
            if (m4 < kmax) {
                consider(dv.x, colbase + j4 * 4 + 0);
                consider(dv.y, colbase + j4 * 4 + 1);
                consider(dv.z, colbase + j4 * 4 + 2);
                consider(dv.w, colbase + j4 * 4 + 3);
            }
        }
    }

    const int q = qblock * QTILE + lane;
    const size_t base = ((size_t)q * NSLICES + slice) * KNN;
    #pragma unroll
    for (int i = 0; i < KNN; ++i) { pdist[base + i] = kd[i]; pidx[base + i] = ki[i]; }
}

// ---------------------------------------------------------------------------
// Kernel 3: merge 32 partial lists per query -> final top-20 + kernel weights
// ---------------------------------------------------------------------------
__global__ void knn_merge_kernel(const float* __restrict__ pdist, const int* __restrict__ pidx,
                                 float* __restrict__ wknn, int* __restrict__ iknn)
{
    int q = blockIdx.x * blockDim.x + threadIdx.x;
    if (q >= BQ) return;

    float kd[KNN]; int ki[KNN];
    #pragma unroll
    for (int i = 0; i < KNN; ++i) { kd[i] = 3.0e38f; ki[i] = 0; }
    float kmax = 3.0e38f; int kpos = 0;

    const size_t base = (size_t)q * NSLICES * KNN;
    for (int c = 0; c < NSLICES * KNN; ++c) {
        float d = pdist[base + c];
        if (d < kmax) {
            int idx = pidx[base + c];
            #pragma unroll
            for (int i = 0; i < KNN; ++i) {
                bool rep = (i == kpos);
                kd[i] = rep ? d   : kd[i];
                ki[i] = rep ? idx : ki[i];
            }
            kmax = kd[0]; kpos = 0;
            #pragma unroll
            for (int i = 1; i < KNN; ++i) { if (kd[i] > kmax) { kmax = kd[i]; kpos = i; } }
        }
    }

    // h = clamped 20th-smallest distance; w_i = exp(-d_i^2 / (2 h^2)), normalized
    float h = fmaxf(sqrtf(fmaxf(kmax, 1e-30f)), 1e-12f);
    float inv2h2 = 1.0f / (2.0f * h * h);
    float w[KNN]; float wsum = 0.f;
    #pragma unroll
    for (int i = 0; i < KNN; ++i) { w[i] = expf(-fmaxf(kd[i], 1e-30f) * inv2h2); wsum += w[i]; }
    float inv = 1.0f / (wsum + 1e-12f);
    #pragma unroll
    for (int i = 0; i < KNN; ++i) { wknn[(size_t)q * KNN + i] = w[i] * inv;
                                    iknn[(size_t)q * KNN + i] = ki[i]; }
}

// ---------------------------------------------------------------------------
// Kernel 4: u_t[q][d] = sum_i w_i * velocity[idx_i][d]
// ---------------------------------------------------------------------------
__global__ void ut_kernel(const float* __restrict__ velocity, const float* __restrict__ wknn,
                          const int* __restrict__ iknn, float* __restrict__ ut)
{
    int t = blockIdx.x * blockDim.x + threadIdx.x;
    if (t >= BQ * DFEAT) return;
    int q = t >> 6, d = t & 63;
    float acc = 0.f;
    #pragma unroll
    for (int i = 0; i < KNN; ++i)
        acc += wknn[(size_t)q * KNN + i] * velocity[(size_t)iknn[(size_t)q * KNN + i] * DFEAT + d];
    ut[t] = acc;
}

// ---------------------------------------------------------------------------
// Kernel 5/6: tiny MLP  x_dot = relu([x,t] @ W1 + b1) @ W2 + b2
// ---------------------------------------------------------------------------
__global__ void mlp1_kernel(const float* __restrict__ z, const float* __restrict__ t,
                            const float* __restrict__ W1, const float* __restrict__ b1,
                            float* __restrict__ hidden)
{
    int i = blockIdx.x * blockDim.x + threadIdx.x;
    if (i >= BQ * HDIM) return;
    int q = i >> 8, h = i & 255;
    const float* xr = z + (size_t)q * (DFEAT + 2);
    float s = b1[h];
    #pragma unroll 8
    for (int j = 0; j < DFEAT; ++j) s += xr[j] * W1[(size_t)j * HDIM + h];
    s += t[0] * W1[(size_t)DFEAT * HDIM + h];
    hidden[i] = fmaxf(s, 0.f);
}

__global__ void mlp2_kernel(const float* __restrict__ hidden, const float* __restrict__ W2,
                            const float* __restrict__ b2, float* __restrict__ xdot)
{
    int i = blockIdx.x * blockDim.x + threadIdx.x;
    if (i >= BQ * DFEAT) return;
    int q = i >> 6, d = i & 63;
    float s = b2[d];
    #pragma unroll 8
    for (int h = 0; h < HDIM; ++h) s += hidden[(size_t)q * HDIM + h] * W2[(size_t)h * DFEAT + d];
    xdot[i] = s;
}

// ---------------------------------------------------------------------------
// Kernel 7: metrics + output assembly: out[q] = [x_dot(64), cos_sim, L2^2]
// ---------------------------------------------------------------------------
__global__ void final_kernel(const float* __restrict__ xdot, const float* __restrict__ ut,
                             float* __restrict__ out)
{
    int q = blockIdx.x * blockDim.x + threadIdx.x;
    if (q >= BQ) return;
    float du = 0.f, nu = 0.f, nx = 0.f, l2 = 0.f;
    #pragma unroll 8
    for (int d = 0; d < DFEAT; ++d) {
        float u  = ut[(size_t)q * DFEAT + d];
        float xd = xdot[(size_t)q * DFEAT + d];
        du += u * xd; nu += u * u; nx += xd * xd;
        float df = u - xd; l2 += df * df;
        out[(size_t)q * 66 + d] = xd;
    }
    float cs = 1.0f - du / (fmaxf(sqrtf(nu), 1e-8f) * fmaxf(sqrtf(nx), 1e-8f));
    out[(size_t)q * 66 + 64] = cs;
    out[(size_t)q * 66 + 65] = l2;
}

// ---------------------------------------------------------------------------
static inline size_t alignUp(size_t x, size_t a) { return (x + a - 1) & ~(a - 1); }

extern "C" void kernel_launch(void* const* d_in, const int* in_sizes, int n_in,
                              void* d_out, int out_size, void* d_ws, size_t ws_size,
                              hipStream_t stream)
{
    const float* t        = (const float*)d_in[0];
    const float* z        = (const float*)d_in[1];
    const float* data     = (const float*)d_in[2];
    const float* velocity = (const float*)d_in[3];
    const float* W1       = (const float*)d_in[4];
    const float* b1       = (const float*)d_in[5];
    const float* W2       = (const float*)d_in[6];
    const float* b2       = (const float*)d_in[7];
    float* out = (float*)d_out;

    char* ws = (char*)d_ws;
    size_t off = 0;
    float* dnorm  = (float*)(ws + off); off = alignUp(off + (size_t)NPTS * 4, 256);
    float* pdist  = (float*)(ws + off); off = alignUp(off + (size_t)BQ * NSLICES * KNN * 4, 256);
    int*   pidx   = (int*)  (ws + off); off = alignUp(off + (size_t)BQ * NSLICES * KNN * 4, 256);
    float* wknn   = (float*)(ws + off); off = alignUp(off + (size_t)BQ * KNN * 4, 256);
    int*   iknn   = (int*)  (ws + off); off = alignUp(off + (size_t)BQ * KNN * 4, 256);
    float* hidden = (float*)(ws + off); off = alignUp(off + (size_t)BQ * HDIM * 4, 256);
    float* xdot   = (float*)(ws + off); off = alignUp(off + (size_t)BQ * DFEAT * 4, 256);
    float* ut     = (float*)(ws + off); off = alignUp(off + (size_t)BQ * DFEAT * 4, 256);
    (void)ws_size; (void)in_sizes; (void)n_in; (void)out_size;

    dnorm_kernel<<<(NPTS + 255) / 256, 256, 0, stream>>>(data, dnorm);
    knn_partial_kernel<<<(BQ / QTILE) * SLICE_GROUPS, 256, 0, stream>>>(z, data, dnorm, pdist, pidx);
    knn_merge_kernel<<<(BQ + 255) / 256, 256, 0, stream>>>(pdist, pidx, wknn, iknn);
    ut_kernel<<<(BQ * DFEAT + 255) / 256, 256, 0, stream>>>(velocity, wknn, iknn, ut);
    mlp1_kernel<<<(BQ * HDIM + 255) / 256, 256, 0, stream>>>(z, t, W1, b1, hidden);
    mlp2_kernel<<<(BQ * DFEAT + 255) / 256, 256, 0, stream>>>(hidden, W2, b2, xdot);
    final_kernel<<<(BQ + 255) / 256, 256, 0, stream>>>(xdot, ut, out);
}